// MarginRankingLoss_74491912782376
// MI455X (gfx1250) — compile-verified
//
#include <hip/hip_runtime.h>
#include <hip/hip_bf16.h>
#include <stdint.h>

// Problem shape (fixed by the reference): x[V=4096, C=128, T=128], target[V, C] int64
#define MARGIN 0.5f
constexpr int   T_DIM            = 128;
constexpr int   ROWS             = 4096 * 128;      // 524288 rows
constexpr int   BLOCKS           = 1024;
constexpr int   THREADS          = 256;             // 8 wave32 per block
constexpr int   WAVES_PER_BLOCK  = THREADS / 32;
constexpr int   TOTAL_WAVES      = BLOCKS * WAVES_PER_BLOCK;   // 8192
constexpr int   ROWS_PER_WAVE    = ROWS / TOTAL_WAVES;         // 64
constexpr int   ROW_BYTES        = T_DIM * 4;       // 512 B = one wave32 b128 transaction
constexpr int   DEPTH            = 6;               // 5 copies in flight -> ~20 MB device-wide,
                                                    // covers ~800ns HBM latency at 23.3 TB/s

typedef float v2f __attribute__((ext_vector_type(2)));
typedef float v8f __attribute__((ext_vector_type(8)));

// CDNA5 async global -> LDS copy, 16 B per lane (one wave = one 512 B row).
// Non-temporal: 256 MiB stream with zero reuse, don't thrash the 192 MB L2.
__device__ __forceinline__ void async_row16(uint32_t lds_off, const void* gsrc) {
    asm volatile("global_load_async_to_lds_b128 %0, %1, off th:TH_LOAD_NT"
                 :: "v"(lds_off), "v"(gsrc)
                 : "memory");
}

__global__ __launch_bounds__(THREADS)
void margin_partial_kernel(const float* __restrict__ x,
                           const long long* __restrict__ tgt,
                           float* __restrict__ partials) {
    // Per-wave DEPTH-slot ring buffer: 8 waves * 6 * 512 B = 24 KiB of the 320 KiB WGP LDS.
    __shared__ __align__(16) float smem[WAVES_PER_BLOCK * DEPTH * T_DIM];
    __shared__ float wave_sums[WAVES_PER_BLOCK];

    const int lane = threadIdx.x & 31;
    const int wid  = threadIdx.x >> 5;
    const int gw   = blockIdx.x * WAVES_PER_BLOCK + wid;   // global wave id

    // 32-bit LDS byte offset of this wave's ring buffer (asm operand only; the
    // generic pointer's low 32 bits are the offset in the group's LDS allocation).
    const uint32_t lds_wave_base = (uint32_t)(uintptr_t)(&smem[0])
                                 + (uint32_t)(wid * (DEPTH * ROW_BYTES));
    const uint32_t lane_slot     = (uint32_t)(lane * 16);

    // ---- pipeline prologue: put DEPTH-1 rows in flight ----
#pragma unroll
    for (int j = 0; j < DEPTH - 1; ++j) {
        const float* g = x + (size_t)(gw + j * TOTAL_WAVES) * T_DIM + lane * 4;
        async_row16(lds_wave_base + (uint32_t)(j * ROW_BYTES) + lane_slot, g);
    }
    int tcur = (int)tgt[gw];

    float acc = 0.0f;
    for (int i = 0; i < ROWS_PER_WAVE; ++i) {
        const int buf = i % DEPTH;

        // Issue copy for row i+DEPTH-1 into the slot freed 1 iteration ago,
        // and prefetch the next row's target while the copy streams.
        int tnext = 0;
        if (i + 1 < ROWS_PER_WAVE) tnext = (int)tgt[gw + (i + 1) * TOTAL_WAVES];
        if (i + DEPTH - 1 < ROWS_PER_WAVE) {
            const int rn = gw + (i + DEPTH - 1) * TOTAL_WAVES;
            const float* gn = x + (size_t)rn * T_DIM + lane * 4;
            async_row16(lds_wave_base + (uint32_t)(((i + DEPTH - 1) % DEPTH) * ROW_BYTES) + lane_slot,
                        gn);
            // Loads complete in order: cnt <= DEPTH-1 means the oldest (row i) landed.
            asm volatile("s_wait_asynccnt 0x5" ::: "memory");
        } else {
            asm volatile("s_wait_asynccnt 0x0" ::: "memory");   // drain tail once, then no-op
        }

        // Direct indexing off the __shared__ array keeps addrspace(3) inference:
        // these must lower to ds_load_b32 / ds_load_b128, NOT flat_load.
        const int row_f = (wid * DEPTH + buf) * T_DIM;

        // Positive score: uniform address -> single broadcast ds_load_b32.
        const float pos = smem[row_f + tcur];

        // Consume current row (ds_load_b128).
        const float4 v = *reinterpret_cast<const float4*>(&smem[row_f + lane * 4]);

        // Hinge over ALL T (target position contributes exactly MARGIN; corrected at the end).
        const float c = MARGIN - pos;
        acc += fmaxf(v.x + c, 0.0f) + fmaxf(v.y + c, 0.0f)
             + fmaxf(v.z + c, 0.0f) + fmaxf(v.w + c, 0.0f);

        tcur = tnext;
    }

    // ---- wave32 reduction via V_WMMA_F32_16X16X4_F32 (EXEC is all ones here) ----
    // A(16x4): A[m,0] = acc[lane m], A[m,2] = acc[lane m+16]; B(4x16) = ones
    //   =>  D[m,n] = acc[m] + acc[m+16], replicated over n.
    v2f a; a[0] = acc;  a[1] = 0.0f;
    v2f b; b[0] = 1.0f; b[1] = 1.0f;
    v8f d = {};
    d = __builtin_amdgcn_wmma_f32_16x16x4_f32(
            /*neg_a=*/false, a, /*neg_b=*/false, b,
            /*c_mod=*/(short)0, d, /*reuse_a=*/false, /*reuse_b=*/false);
    // Lanes 0-15 hold D rows 0..7 in VGPR0..7, lanes 16-31 hold rows 8..15.
    float s = d[0] + d[1] + d[2] + d[3] + d[4] + d[5] + d[6] + d[7];
    s += __shfl_xor(s, 16, 32);          // full wave total in every lane

    if (lane == 0) wave_sums[wid] = s;
    __syncthreads();
    if (threadIdx.x == 0) {
        float bs = 0.0f;
#pragma unroll
        for (int w = 0; w < WAVES_PER_BLOCK; ++w) bs += wave_sums[w];
        partials[blockIdx.x] = bs;       // deterministic: one slot per block, no atomics
    }
}

__global__ __launch_bounds__(256)
void margin_final_kernel(const float* __restrict__ partials, float* __restrict__ out) {
    __shared__ float sm[256];
    float s = 0.0f;
    for (int i = threadIdx.x; i < BLOCKS; i += 256) s += partials[i];
    sm[threadIdx.x] = s;
    __syncthreads();
    for (int st = 128; st > 0; st >>= 1) {
        if (threadIdx.x < st) sm[threadIdx.x] += sm[threadIdx.x + st];
        __syncthreads();
    }
    if (threadIdx.x == 0) {
        // Remove the target-position contribution (MARGIN per row), then average.
        const double total_excl = (double)sm[0] - (double)ROWS * (double)MARGIN;
        const double cnt        = (double)ROWS * (double)(T_DIM - 1);
        out[0] = (float)(total_excl / cnt);
    }
}

extern "C" void kernel_launch(void* const* d_in, const int* in_sizes, int n_in,
                              void* d_out, int out_size, void* d_ws, size_t ws_size,
                              hipStream_t stream) {
    const float*     x   = (const float*)d_in[0];
    const long long* tgt = (const long long*)d_in[1];   // int64 targets
    float* partials = (float*)d_ws;                     // BLOCKS floats = 4 KiB scratch
    float* out      = (float*)d_out;

    margin_partial_kernel<<<BLOCKS, THREADS, 0, stream>>>(x, tgt, partials);
    margin_final_kernel<<<1, 256, 0, stream>>>(partials, out);
}